// ROIPooling_54262616818221
// MI455X (gfx1250) — compile-verified
//
#include <hip/hip_runtime.h>
#include <cfloat>

// ROI max-pooling for MI455X (gfx1250).
// features: (2, 256, 50, 50) f32, boxes: (2, 64, 4) f32, image_size: int scalar
// out: (128, 256, 7, 7) f32
//
// ~15 MB total traffic -> <1us at 23.3 TB/s; latency-bound. CDNA5 path:
// one Tensor Data Mover descriptor per block DMAs the (rw x rh x 32ch) ROI
// tile into LDS (TENSORcnt), replacing ~200 per-lane loads + their integer
// division address math. TDM issued via inline asm (portable across the
// 5-arg/6-arg builtin split between toolchains).

#define USE_TDM 1

#define RS 7          // output grid S
#define FH 50
#define FW 50
#define NCH 256
#define NB_PER_B 64   // boxes per batch image
#define CHB 32        // channels per block
#define SLOT 224      // max ROI floats per channel (ROI <= 14x14 = 196)
#define BLOCK 256

typedef unsigned int u32;
typedef u32 __attribute__((ext_vector_type(4))) u32x4;
typedef int __attribute__((ext_vector_type(4))) i32x4;
typedef int __attribute__((ext_vector_type(8))) i32x8;

__global__ __launch_bounds__(BLOCK) void roi_pool_tdm_kernel(
    const float* __restrict__ features, const float* __restrict__ boxes,
    const int* __restrict__ img, float* __restrict__ out)
{
  const int box   = blockIdx.x;          // 0..127
  const int c0    = blockIdx.y * CHB;    // channel group base
  const int tid   = threadIdx.x;
  const int batch = box / NB_PER_B;

  __shared__ float smem[CHB * SLOT];     // 28 KB

  const float img_f   = (float)img[0];
  const float scale_h = (float)FH / img_f;
  const float scale_w = (float)FW / img_f;

  const float bx = boxes[box * 4 + 0];
  const float by = boxes[box * 4 + 1];
  const float bw = boxes[box * 4 + 2];
  const float bh = boxes[box * 4 + 3];

  // jnp.round == round-half-even == rintf (v_rndne_f32)
  int x1 = (int)rintf(bx * scale_w);            x1 = x1 > 0 ? x1 : 0;
  int y1 = (int)rintf(by * scale_h);            y1 = y1 > 0 ? y1 : 0;
  int x2 = (int)rintf((bx + bw) * scale_w) + 1; x2 = x2 < FW ? x2 : FW;
  int y2 = (int)rintf((by + bh) * scale_h) + 1; y2 = y2 < FH ? y2 : FH;
  if (x2 <= x1) x2 = (x1 + 1 < FW) ? x1 + 1 : FW;
  if (y2 <= y1) y2 = (y1 + 1 < FH) ? y1 + 1 : FH;
  const int rh   = y2 - y1;
  const int rw   = x2 - x1;
  const int area = rh * rw;

  const float* fbase =
      features + (((size_t)(batch * NCH + c0)) * FH + y1) * FW + x1;

  const bool staged = (area > 0) && (area <= SLOT);   // uniform per block
  if (staged) {
#if USE_TDM
    // --- Tensor Data Mover: one descriptor loads the whole 3D tile ---
    // tile: dim0 = rw elems (contig), dim1 = rh rows (stride 50),
    //       dim2 = CHB channels (stride 2500). data_size = 4B.
    // LDS result: packed rw*rh floats per channel, channels consecutive.
    if (tid < warpSize) {   // one wave issues the DMA (TDM ignores EXEC)
      const u32 rwu = (u32)__builtin_amdgcn_readfirstlane(rw);
      const u32 rhu = (u32)__builtin_amdgcn_readfirstlane(rh);
      const u32 glo = (u32)__builtin_amdgcn_readfirstlane(
          (int)(u32)(size_t)fbase);
      const u32 ghi = (u32)__builtin_amdgcn_readfirstlane(
          (int)(u32)(((size_t)fbase) >> 32));
      const u32 lds_base = (u32)(size_t)(&smem[0]);  // low 32 bits = LDS offset
      const u32 TDIM = 0x40000000u;                  // huge dims: no OOB zeroing

      u32x4 g0;
      g0.x = 1u;                                   // count=1 (valid), load, user
      g0.y = lds_base;                             // lds_addr (bytes)
      g0.z = glo;                                  // global_addr[31:0]
      g0.w = (ghi & 0x01FFFFFFu) | (2u << 30);     // global_addr[56:32] | type=2

      i32x8 g1;
      g1.s0 = (int)(2u << 16);                     // wg_mask=0 | data_size=4B
      g1.s1 = (int)((TDIM & 0xFFFFu) << 16);       // tensor_dim0[15:0]<<16
      g1.s2 = (int)((TDIM >> 16) | ((TDIM & 0xFFFFu) << 16)); // td0 hi | td1 lo
      g1.s3 = (int)((TDIM >> 16) | (rwu << 16));   // td1 hi | tile_dim0=rw
      g1.s4 = (int)(rhu | ((u32)CHB << 16));       // tile_dim1=rh | tile_dim2=32
      g1.s5 = (int)(FW);                           // tensor_dim0_stride = 50
      g1.s6 = (int)(((u32)(FH * FW) & 0xFFFFu) << 16); // s0 hi=0 | s1 lo=2500
      g1.s7 = (int)((u32)(FH * FW) >> 16);         // tensor_dim1_stride hi

      i32x4 g2;
      g2.x = (int)TDIM;                            // tensor_dim2 (no OOB)
      g2.y = 0;                                    // tensor_dim3 (unused)
      g2.z = 0;                                    // tensor_dim2_stride lo
      g2.w = 0;                                    // stride hi | tile_dim3=0
      i32x4 g3 = {0, 0, 0, 0};

      asm volatile("tensor_load_to_lds %0, %1, %2, %3"
                   :: "s"(g0), "s"(g1), "s"(g2), "s"(g3)
                   : "memory");
      __builtin_amdgcn_s_wait_tensorcnt(0);
    }
#else
    // --- per-lane async gather fallback ---
    const int total  = CHB * area;
    const int padded = (total + BLOCK - 1) & ~(BLOCK - 1);
    for (int t = tid; t < padded; t += BLOCK) {
      int tt  = t < total ? t : total - 1;
      int ch  = tt / area;
      int rem = tt - ch * area;
      int r   = rem / rw;
      int c   = rem - r * rw;
      const float* g = fbase + (size_t)ch * (FH * FW) + r * FW + c;
      unsigned lds = (unsigned)(size_t)(&smem[ch * area + rem]);
      asm volatile("global_load_async_to_lds_b32 %0, %1, off"
                   :: "v"(lds), "v"(g)
                   : "memory");
    }
    asm volatile("s_wait_asynccnt 0" ::: "memory");
#endif
    __syncthreads();
  }

  // Each thread computes output bins: 32 ch * 49 bins = 1568 per block.
  for (int idx = tid; idx < CHB * RS * RS; idx += BLOCK) {
    int ch  = idx / (RS * RS);
    int bin = idx - ch * (RS * RS);
    int i   = bin / RS;
    int j   = bin - i * RS;
    int hs  = (i * rh) / RS;
    int he  = ((i + 1) * rh + RS - 1) / RS;
    int ws  = (j * rw) / RS;
    int we  = ((j + 1) * rw + RS - 1) / RS;
    float m = -FLT_MAX;
    if (staged) {
      const float* sp = &smem[ch * area];          // TDM packs rw*rh per channel
      for (int r = hs; r < he; ++r)
        for (int c = ws; c < we; ++c)
          m = fmaxf(m, sp[r * rw + c]);
    } else {
      const float* gp = fbase + (size_t)ch * (FH * FW);
      for (int r = hs; r < he; ++r)
        for (int c = ws; c < we; ++c)
          m = fmaxf(m, gp[r * FW + c]);
    }
    out[(((size_t)box * NCH + (c0 + ch)) * RS + i) * RS + j] = m;
  }
}

extern "C" void kernel_launch(void* const* d_in, const int* in_sizes, int n_in,
                              void* d_out, int out_size, void* d_ws, size_t ws_size,
                              hipStream_t stream) {
  const float* features = (const float*)d_in[0];
  const float* boxes    = (const float*)d_in[1];
  const int*   img      = (const int*)d_in[2];
  float*       out      = (float*)d_out;

  const int n_boxes = in_sizes[1] / 4;          // 128
  dim3 grid(n_boxes, NCH / CHB);                // 128 x 8 blocks
  roi_pool_tdm_kernel<<<grid, BLOCK, 0, stream>>>(features, boxes, img, out);
}